// KMeans_9921374454451
// MI455X (gfx1250) — compile-verified
//
#include <hip/hip_runtime.h>

typedef __attribute__((ext_vector_type(2))) float v2f;
typedef __attribute__((ext_vector_type(4))) float v4f;
typedef __attribute__((ext_vector_type(8))) float v8f;
typedef __attribute__((ext_vector_type(4))) unsigned int u32x4;
typedef __attribute__((ext_vector_type(4))) int i32x4;
typedef __attribute__((ext_vector_type(8))) int i32x8;

#define NPTS    262144
#define KCENT   512
#define DIM     64
#define LDSSTR  68          // padded LDS row stride in floats (64 data + 4 pad)
#define ROWS_PB 256         // 8 waves x two 16-row M tiles
#define THREADS 256
#define NTILES  (KCENT / 16)

#define HAVE_TDM __has_builtin(__builtin_amdgcn_tensor_load_to_lds)

__global__ __launch_bounds__(THREADS)
void kmeans_assign_wmma(const float* __restrict__ x,
                        const float* __restrict__ cent,
                        float* __restrict__ out)   // [KCENT*DIM centroids | NPTS assignments]
{
    extern __shared__ float smem[];                 // KCENT*LDSSTR padded centroid tile
    float* csq = smem + KCENT * LDSSTR;             // + KCENT squared norms

    const int tid  = threadIdx.x;
    const int lane = tid & 31;
    const int wave = tid >> 5;
    const int half = lane >> 4;                     // K half-select (wave32 halves)
    const int mn   = lane & 15;                     // M for A-frag / N for B,C-frags

    // ---- pass-through output 0: centroids ----
    {
        int cidx = blockIdx.x * THREADS + tid;
        if (cidx < KCENT * DIM) out[cidx] = cent[cidx];
    }

#if HAVE_TDM
    // ---- TDM: DMA centroids[512x64] f32 -> LDS with in-flight pad to stride 68 ----
    if (wave == 0) {
        unsigned lds_base = (unsigned)(unsigned long long)
            (__attribute__((address_space(3))) float*)smem;
        unsigned long long ga = (unsigned long long)cent;
        // D# group0: count=1 (valid), lds_addr, 57-bit global addr, type=2
        u32x4 g0 = { 1u,
                     lds_base,
                     (unsigned)ga,
                     (unsigned)((ga >> 32) & 0x01FFFFFFu) | (2u << 30) };
        // D# group1: data_size=4B(2), pad_enable, pad_interval=64dw(5), pad_amount=4dw(3)
        //            tensor_dim0=64, tensor_dim1=512, tile_dim0=64, tile_dim1=512,
        //            tensor_dim0_stride=64
        i32x8 g1 = { (int)((2u << 16) | (1u << 20) | (5u << 22) | (3u << 25)),
                     (int)(DIM   << 16),          // tensor_dim0[15:0]
                     (int)(KCENT << 16),          // tensor_dim1[15:0]
                     (int)(DIM   << 16),          // tile_dim0
                     (int)KCENT,                  // tile_dim1
                     (int)DIM,                    // tensor_dim0_stride[31:0]
                     0, 0 };
        i32x4 gz = { 0, 0, 0, 0 };
  #if __has_include(<hip/amd_detail/amd_gfx1250_TDM.h>)
        i32x8 gz8 = { 0, 0, 0, 0, 0, 0, 0, 0 };
        __builtin_amdgcn_tensor_load_to_lds(g0, g1, gz, gz, gz8, 0);
  #else
        __builtin_amdgcn_tensor_load_to_lds(g0, g1, gz, gz, 0);
  #endif
    }
#else
    // ---- fallback: synchronous staged copy into padded LDS ----
    #pragma unroll
    for (int i = 0; i < (KCENT * DIM) / (THREADS * 4); ++i) {
        int f   = (tid + i * THREADS) * 4;
        int row = f >> 6;
        int col = f & 63;
        *(v4f*)(smem + row * LDSSTR + col) = *(const v4f*)(cent + f);
    }
#endif

    // ---- preload A fragments for two 16-row M tiles (overlaps the TDM DMA) ----
    // ISA 16x4 f32 A layout: lane L<16 -> row M=L, K=4c+{0,1}; lane L>=16 -> M=L-16, K=4c+{2,3}
    const int row_base = blockIdx.x * ROWS_PB + wave * 32;
    const float* xa0 = x + (row_base + mn) * DIM + 2 * half;
    const float* xa1 = xa0 + 16 * DIM;
    v2f a0[16], a1[16];
    #pragma unroll
    for (int kc = 0; kc < 16; ++kc) {
        a0[kc] = *(const v2f*)(xa0 + kc * 4);
        a1[kc] = *(const v2f*)(xa1 + kc * 4);
    }

#if HAVE_TDM
    if (wave == 0) __builtin_amdgcn_s_wait_tensorcnt(0);   // wave0's DMA done before barrier signal
#endif
    __syncthreads();

    // ---- per-centroid squared norms ----
    #pragma unroll
    for (int i = 0; i < KCENT / THREADS; ++i) {
        int k = tid + i * THREADS;
        const float* row = smem + k * LDSSTR;
        float s = 0.f;
        #pragma unroll
        for (int d = 0; d < DIM; ++d) s = fmaf(row[d], row[d], s);
        csq[k] = s;
    }
    __syncthreads();

    float best0[8], best1[8];
    int   bi0[8],   bi1[8];
    #pragma unroll
    for (int r = 0; r < 8; ++r) {
        best0[r] = __builtin_inff(); bi0[r] = 0;
        best1[r] = __builtin_inff(); bi1[r] = 0;
    }

    // ---- sweep 32 N-tiles; each B fragment feeds TWO WMMAs (independent acc chains) ----
    for (int t = 0; t < NTILES; ++t) {
        v8f acc0 = {}, acc1 = {};
        const float* bb = smem + (t * 16 + mn) * LDSSTR + 2 * half;
        #pragma unroll
        for (int kc = 0; kc < 16; ++kc) {
            v2f b = *(const v2f*)(bb + kc * 4);
            acc0 = __builtin_amdgcn_wmma_f32_16x16x4_f32(
                false, a0[kc], false, b, (short)0, acc0, false, false);
            acc1 = __builtin_amdgcn_wmma_f32_16x16x4_f32(
                false, a1[kc], false, b, (short)0, acc1, false, false);
        }
        // score = c_sq[n] - 2*cross   (x_sq constant per row drops out of argmin)
        float cs     = csq[t * 16 + mn];
        int   n_glob = t * 16 + mn;
        #pragma unroll
        for (int r = 0; r < 8; ++r) {
            float s0 = fmaf(-2.f, acc0[r], cs);
            float s1 = fmaf(-2.f, acc1[r], cs);
            if (s0 < best0[r]) { best0[r] = s0; bi0[r] = n_glob; }
            if (s1 < best1[r]) { best1[r] = s1; bi1[r] = n_glob; }
        }
    }

    // ---- argmin across the 16 lanes of each M-row (xor masks stay inside halves) ----
    float* assign_out = out + KCENT * DIM;
    #pragma unroll
    for (int g = 0; g < 2; ++g) {
        float* best = g ? best1 : best0;
        int*   bidx = g ? bi1   : bi0;
        int    rb   = row_base + g * 16;
        #pragma unroll
        for (int r = 0; r < 8; ++r) {
            float s = best[r];
            int   i = bidx[r];
            #pragma unroll
            for (int m = 8; m >= 1; m >>= 1) {
                float so = __shfl_xor(s, m, 32);
                int   io = __shfl_xor(i, m, 32);
                if (so < s || (so == s && io < i)) { s = so; i = io; }
            }
            if (lane == 0)       assign_out[rb + r]     = (float)i;   // M = r
            else if (lane == 16) assign_out[rb + 8 + r] = (float)i;   // M = r+8
        }
    }
}

extern "C" void kernel_launch(void* const* d_in, const int* in_sizes, int n_in,
                              void* d_out, int out_size, void* d_ws, size_t ws_size,
                              hipStream_t stream) {
    (void)in_sizes; (void)n_in; (void)out_size; (void)d_ws; (void)ws_size;
    const float* x    = (const float*)d_in[0];   // [262144, 64]
    const float* cent = (const float*)d_in[1];   // [512, 64]
    float*       out  = (float*)d_out;           // [512*64 centroids | 262144 assignments]

    const int blocks  = NPTS / ROWS_PB;                           // 1024
    const size_t lds  = (KCENT * LDSSTR + KCENT) * sizeof(float); // ~138 KB dynamic LDS
    kmeans_assign_wmma<<<blocks, THREADS, lds, stream>>>(x, cent, out);
}